// HelicalGNNFrontend_14757507629325
// MI455X (gfx1250) — compile-verified
//
#include <hip/hip_runtime.h>
#include <stdint.h>

#define HID 128
#define EDIM 32
#define NEG_SLOPE 0.2f

typedef float v2f __attribute__((ext_vector_type(2)));
typedef float v8f __attribute__((ext_vector_type(8)));

// ---------------------------------------------------------------------------
// CDNA5 f32 WMMA: D(16x16,f32) = A(16x4,f32) * B(4x16,f32) + C
// A frag: lane<16 holds (M=lane, K=0..1), lane>=16 holds (M=lane-16, K=2..3)
// B frag: v0 = rows K=0 (lanes0-15) / K=2 (lanes16-31); v1 = K=1 / K=3
// C/D   : VGPR v -> M = v + 8*(lane/16), N = lane%16
// ---------------------------------------------------------------------------
__device__ __forceinline__ v8f wmma4(v2f a, v2f b, v8f c) {
  return __builtin_amdgcn_wmma_f32_16x16x4_f32(false, a, false, b, (short)0, c,
                                               false, false);
}

__device__ __forceinline__ void atomicMaxFloat(float* addr, float val) {
  if (val >= 0.0f)
    atomicMax((int*)addr, __float_as_int(val));
  else
    atomicMin((unsigned int*)addr, (unsigned int)__float_as_int(val));
}

// ---------------------------- elementwise utils ----------------------------
__global__ void fill_f32(float* __restrict__ p, float v, long long n) {
  long long i = (long long)blockIdx.x * blockDim.x + threadIdx.x;
  if (i < n) p[i] = v;
}

// degree + sum of incoming edge_attr; one thread per (edge, col)
__global__ void deg_loopattr(const long long* __restrict__ dst,
                             const float* __restrict__ ea,
                             float* __restrict__ deg, float* __restrict__ la,
                             long long E) {
  long long i = (long long)blockIdx.x * blockDim.x + threadIdx.x;
  if (i >= E * EDIM) return;
  long long e = i >> 5;            // EDIM == 32
  int c = (int)(i & (EDIM - 1));
  long long d = dst[e];
  if (c == 0) atomicAdd(&deg[d], 1.0f);
  atomicAdd(&la[d * EDIM + c], ea[i]);
}

__global__ void loopattr_norm(float* __restrict__ la,
                              const float* __restrict__ deg, long long N) {
  long long i = (long long)blockIdx.x * blockDim.x + threadIdx.x;
  if (i >= N * EDIM) return;
  la[i] = la[i] / fmaxf(deg[i >> 5], 1.0f);
}

// ---------------------------------------------------------------------------
// out[M x 128] = A[M x 128] @ W[128 x 128] (+ bias)
// block = 256 threads = 8 waves; wave w owns output col tile w (16 cols);
// grid.x = M/16 row tiles. W staged in 64KB LDS. M must be a multiple of 16.
// ---------------------------------------------------------------------------
__global__ void gemm128(const float* __restrict__ A, const float* __restrict__ W,
                        const float* __restrict__ bias, float* __restrict__ out,
                        int M) {
  __shared__ float lW[HID * HID];
  for (int i = threadIdx.x; i < HID * HID; i += blockDim.x) lW[i] = W[i];
  __syncthreads();
  int wave = threadIdx.x >> 5;
  int lane = threadIdx.x & 31;
  int half = lane >> 4;
  int l16 = lane & 15;
  int row0 = blockIdx.x * 16;
  if (row0 >= M) return;
  int col = wave * 16 + l16;
  const float* arow = A + (size_t)(row0 + l16) * HID;
  v8f c = {0.f, 0.f, 0.f, 0.f, 0.f, 0.f, 0.f, 0.f};
#pragma unroll
  for (int k = 0; k < HID; k += 4) {
    v2f a, b;
    a.x = arow[k + 2 * half];
    a.y = arow[k + 2 * half + 1];
    b.x = lW[(k + 2 * half) * HID + col];
    b.y = lW[(k + 2 * half + 1) * HID + col];
    c = wmma4(a, b, c);
  }
  float bv = bias ? bias[col] : 0.0f;
#pragma unroll
  for (int v = 0; v < 8; ++v)
    out[(size_t)(row0 + v + 8 * half) * HID + col] = c[v] + bv;
}

// ---------------------------------------------------------------------------
// Per-edge GATv2 logit, 16 edges per wave via WMMA:
//   msg = ea@We + xl[src] + xr[dst];  logit = att . leaky_relu(msg)
// Also atomic-max into mbuf[dst]. Edges e >= E are self-loops (src=dst=e-E,
// attr = loop_attr).  block = 256 = 8 waves, We staged in LDS (16KB).
// ---------------------------------------------------------------------------
__global__ void edge_logits(const long long* __restrict__ srcIdx,
                            const long long* __restrict__ dstIdx,
                            const float* __restrict__ eattr,
                            const float* __restrict__ la,
                            const float* __restrict__ xl,
                            const float* __restrict__ xr,
                            const float* __restrict__ We,   // [32 x 128]
                            const float* __restrict__ att,  // [128]
                            float* __restrict__ logit, float* __restrict__ mbuf,
                            long long E, long long E2) {
  __shared__ float lWe[EDIM * HID];
  for (int i = threadIdx.x; i < EDIM * HID; i += blockDim.x) lWe[i] = We[i];
  __syncthreads();
  int wave = threadIdx.x >> 5;
  int lane = threadIdx.x & 31;
  int half = lane >> 4;
  int l16 = lane & 15;
  long long e0 = ((long long)blockIdx.x * 8 + wave) * 16;
  if (e0 >= E2) return;

  // A fragments: this lane supplies edge row M = l16 (K split by lane half)
  long long eA = e0 + l16;
  if (eA >= E2) eA = E2 - 1;
  const float* earow =
      (eA < E) ? (eattr + (size_t)eA * EDIM) : (la + (size_t)(eA - E) * EDIM);
  v2f afrag[8];
#pragma unroll
  for (int k = 0; k < 8; ++k) {
    afrag[k].x = earow[4 * k + 2 * half];
    afrag[k].y = earow[4 * k + 2 * half + 1];
  }
  // src/dst for the C rows this lane owns: M = 8*half + v
  long long sA[8], dA[8];
#pragma unroll
  for (int v = 0; v < 8; ++v) {
    long long e = e0 + 8 * half + v;
    if (e >= E2) e = E2 - 1;
    if (e < E) {
      sA[v] = srcIdx[e];
      dA[v] = dstIdx[e];
    } else {
      sA[v] = dA[v] = e - E;
    }
  }
  float acc[8] = {0, 0, 0, 0, 0, 0, 0, 0};
  for (int t = 0; t < 8; ++t) {  // 8 col tiles of 16 -> full 128 cols
    int col = t * 16 + l16;
    v8f c = {0.f, 0.f, 0.f, 0.f, 0.f, 0.f, 0.f, 0.f};
#pragma unroll
    for (int k = 0; k < 8; ++k) {  // K = 32 in steps of 4
      v2f b;
      b.x = lWe[(4 * k + 2 * half) * HID + col];
      b.y = lWe[(4 * k + 2 * half + 1) * HID + col];
      c = wmma4(afrag[k], b, c);
    }
    float av = att[col];
#pragma unroll
    for (int v = 0; v < 8; ++v) {
      float msg = c[v] + xl[(size_t)sA[v] * HID + col] +
                  xr[(size_t)dA[v] * HID + col];
      float lr = msg > 0.0f ? msg : NEG_SLOPE * msg;
      acc[v] += lr * av;
    }
  }
  // reduce partial dot across the 16 lanes of each half
#pragma unroll
  for (int v = 0; v < 8; ++v) {
    float a = acc[v];
    a += __shfl_xor(a, 1, 16);
    a += __shfl_xor(a, 2, 16);
    a += __shfl_xor(a, 4, 16);
    a += __shfl_xor(a, 8, 16);
    acc[v] = a;
  }
  if (l16 == 0) {
#pragma unroll
    for (int v = 0; v < 8; ++v) {
      long long e = e0 + 8 * half + v;
      if (e < E2) {
        logit[e] = acc[v];
        atomicMaxFloat(&mbuf[dA[v]], acc[v]);
      }
    }
  }
}

__global__ void softmax_ez(const long long* __restrict__ dstIdx,
                           const float* __restrict__ logit,
                           const float* __restrict__ mbuf,
                           float* __restrict__ ez, float* __restrict__ den,
                           long long E, long long E2) {
  long long e = (long long)blockIdx.x * blockDim.x + threadIdx.x;
  if (e >= E2) return;
  long long d = (e < E) ? dstIdx[e] : (e - E);
  float z = __expf(logit[e] - mbuf[d]);
  ez[e] = z;
  atomicAdd(&den[d], z);
}

// agg[dst] += (ez/den[dst]) * xl[src]; one thread per (edge, 4 cols)
__global__ void aggregate(const long long* __restrict__ srcIdx,
                          const long long* __restrict__ dstIdx,
                          const float* __restrict__ xl,
                          const float* __restrict__ ez,
                          const float* __restrict__ den,
                          float* __restrict__ agg, long long E, long long E2) {
  long long i = (long long)blockIdx.x * blockDim.x + threadIdx.x;
  if (i >= E2 * (HID / 4)) return;
  long long e = i >> 5;  // HID/4 == 32
  int c4 = (int)(i & 31);
  long long s, d;
  if (e < E) {
    s = srcIdx[e];
    d = dstIdx[e];
  } else {
    s = d = e - E;
  }
  float alpha = ez[e] / den[d];
  const float4 xv = ((const float4*)(xl + (size_t)s * HID))[c4];
  float* o = agg + (size_t)d * HID + c4 * 4;
  atomicAdd(o + 0, alpha * xv.x);
  atomicAdd(o + 1, alpha * xv.y);
  atomicAdd(o + 2, alpha * xv.z);
  atomicAdd(o + 3, alpha * xv.w);
}

__global__ void bias_silu(const float* __restrict__ agg,
                          const float* __restrict__ bo, float* __restrict__ out,
                          long long total) {
  long long i = (long long)blockIdx.x * blockDim.x + threadIdx.x;
  if (i >= total) return;
  float v = agg[i] + bo[i & (HID - 1)];
  out[i] = v / (1.0f + __expf(-v));
}

// ---------------------------------------------------------------------------
static inline unsigned cdiv_u(long long a, long long b) {
  return (unsigned)((a + b - 1) / b);
}

static void gat_layer(const float* h, const long long* eidx,
                      const float* eattr, const float* la, const float* Wl,
                      const float* bl, const float* Wr, const float* We,
                      const float* att, const float* bo, float* xl, float* xr,
                      float* mbuf, float* den, float* lg, float* ez, float* agg,
                      float* hout, long long N, long long E,
                      hipStream_t stream) {
  long long E2 = E + N;
  gemm128<<<cdiv_u(N, 16), 256, 0, stream>>>(h, Wl, bl, xl, (int)N);
  gemm128<<<cdiv_u(N, 16), 256, 0, stream>>>(h, Wr, nullptr, xr, (int)N);
  fill_f32<<<cdiv_u(N, 256), 256, 0, stream>>>(mbuf, -__builtin_inff(), N);
  fill_f32<<<cdiv_u(N, 256), 256, 0, stream>>>(den, 0.0f, N);
  fill_f32<<<cdiv_u(N * HID, 256), 256, 0, stream>>>(agg, 0.0f, N * HID);
  long long waves = (E2 + 15) / 16;
  edge_logits<<<cdiv_u(waves, 8), 256, 0, stream>>>(
      eidx, eidx + E, eattr, la, xl, xr, We, att, lg, mbuf, E, E2);
  softmax_ez<<<cdiv_u(E2, 256), 256, 0, stream>>>(eidx + E, lg, mbuf, ez, den,
                                                  E, E2);
  aggregate<<<cdiv_u(E2 * 32, 256), 256, 0, stream>>>(eidx, eidx + E, xl, ez,
                                                      den, agg, E, E2);
  bias_silu<<<cdiv_u(N * HID, 256), 256, 0, stream>>>(agg, bo, hout, N * HID);
}

extern "C" void kernel_launch(void* const* d_in, const int* in_sizes, int n_in,
                              void* d_out, int out_size, void* d_ws,
                              size_t ws_size, hipStream_t stream) {
  (void)n_in; (void)out_size; (void)ws_size;
  const float* x = (const float*)d_in[0];
  const long long* eidx = (const long long*)d_in[1];  // [2 x E] int64
  const float* eattr = (const float*)d_in[2];
  const float* W_emb = (const float*)d_in[3];
  const float* b_emb = (const float*)d_in[4];
  const float* Wl1 = (const float*)d_in[5];
  const float* bl1 = (const float*)d_in[6];
  const float* Wr1 = (const float*)d_in[7];
  const float* We1 = (const float*)d_in[8];
  const float* att1 = (const float*)d_in[9];
  const float* bo1 = (const float*)d_in[10];
  const float* Wl2 = (const float*)d_in[11];
  const float* bl2 = (const float*)d_in[12];
  const float* Wr2 = (const float*)d_in[13];
  const float* We2 = (const float*)d_in[14];
  const float* att2 = (const float*)d_in[15];
  const float* bo2 = (const float*)d_in[16];

  long long N = in_sizes[0] / HID;
  long long E = in_sizes[2] / EDIM;
  long long E2 = E + N;

  float* ws = (float*)d_ws;
  float* h = ws;    ws += N * HID;
  float* agg = ws;  ws += N * HID;
  float* xl = ws;   ws += N * HID;
  float* xr = ws;   ws += N * HID;
  float* la = ws;   ws += N * EDIM;
  float* deg = ws;  ws += N;
  float* mbuf = ws; ws += N;
  float* den = ws;  ws += N;
  float* lg = ws;   ws += E2;
  float* ez = ws;   ws += E2;

  // self-loop attr = mean incoming edge_attr (layer independent)
  fill_f32<<<cdiv_u(N, 256), 256, 0, stream>>>(deg, 0.0f, N);
  fill_f32<<<cdiv_u(N * EDIM, 256), 256, 0, stream>>>(la, 0.0f, N * EDIM);
  deg_loopattr<<<cdiv_u(E * EDIM, 256), 256, 0, stream>>>(eidx + E, eattr, deg,
                                                          la, E);
  loopattr_norm<<<cdiv_u(N * EDIM, 256), 256, 0, stream>>>(la, deg, N);

  // h = x @ W_emb + b_emb
  gemm128<<<cdiv_u(N, 16), 256, 0, stream>>>(x, W_emb, b_emb, h, (int)N);

  // layer 1 (writes silu result back into h)
  gat_layer(h, eidx, eattr, la, Wl1, bl1, Wr1, We1, att1, bo1, xl, xr, mbuf,
            den, lg, ez, agg, h, N, E, stream);
  // layer 2 -> final output
  gat_layer(h, eidx, eattr, la, Wl2, bl2, Wr2, We2, att2, bo2, xl, xr, mbuf,
            den, lg, ez, agg, (float*)d_out, N, E, stream);
}